// GreedyGroupedRouter_27273042330076
// MI455X (gfx1250) — compile-verified
//
#include <hip/hip_runtime.h>

#define N_EXPERTS 256
#define N_GROUPS 8
#define GROUP_SIZE 32
#define TOP_K 8

#define WAVES_PER_BLOCK 8
#define THREADS (WAVES_PER_BLOCK * 32)
#define NBLOCKS 2048
#define NBUF 3

typedef float v4f __attribute__((ext_vector_type(4)));
typedef int   v4i __attribute__((ext_vector_type(4)));

#if defined(__gfx1250__) && \
    __has_builtin(__builtin_amdgcn_global_load_async_to_lds_b128) && \
    __has_builtin(__builtin_amdgcn_s_wait_asynccnt)
#define USE_ASYNC_LDS 1
typedef __attribute__((address_space(1))) v4i gv4i;  // global (addrspace 1) int4
typedef __attribute__((address_space(3))) v4i lv4i;  // LDS (addrspace 3) int4
#else
#define USE_ASYNC_LDS 0
#endif

__global__ void zero_hist_kernel(float* __restrict__ hist) {
    if (threadIdx.x < N_EXPERTS) hist[threadIdx.x] = 0.0f;
}

#if USE_ASYNC_LDS
__device__ __forceinline__ void prefetch_row(const float* __restrict__ logits,
                                             int row, float* __restrict__ dst,
                                             int lane) {
    const float* src = logits + (size_t)row * N_EXPERTS;
    __builtin_amdgcn_global_load_async_to_lds_b128(
        (gv4i*)(src + lane * 4), (lv4i*)(dst + lane * 4), 0, 0);
    __builtin_amdgcn_global_load_async_to_lds_b128(
        (gv4i*)(src + 128 + lane * 4), (lv4i*)(dst + 128 + lane * 4), 0, 0);
}
#endif

__global__ __launch_bounds__(THREADS) void router_kernel(
    const float* __restrict__ logits, int S,
    float* __restrict__ out_logits,
    float* __restrict__ out_rw,
    float* __restrict__ out_tw,
    float* __restrict__ out_tid,
    float* __restrict__ out_hist)
{
    __shared__ unsigned int hist[N_EXPERTS];
#if USE_ASYNC_LDS
    __shared__ __align__(16) float stage[WAVES_PER_BLOCK][NBUF][N_EXPERTS];
#endif
    for (int i = threadIdx.x; i < N_EXPERTS; i += THREADS) hist[i] = 0u;
    __syncthreads();

    const int lane = threadIdx.x & 31;
    const int wib  = threadIdx.x >> 5;
    const int gw   = blockIdx.x * WAVES_PER_BLOCK + wib;
    const int nw   = gridDim.x * WAVES_PER_BLOCK;

#if USE_ASYNC_LDS
    int pb = 0;
    if (gw < S)      prefetch_row(logits, gw,      &stage[wib][0][0], lane);
    if (gw + nw < S) prefetch_row(logits, gw + nw, &stage[wib][1][0], lane);
#endif

    for (int t = gw; t < S; t += nw) {
        v4f a, b;
#if USE_ASYNC_LDS
        // 3-stage pipeline: keep 2 rows (4 async ops) in flight per wave.
        if (t + 2 * nw < S) {
            prefetch_row(logits, t + 2 * nw, &stage[wib][(pb + 2) % NBUF][0], lane);
            __builtin_amdgcn_s_wait_asynccnt(4);  // rows t+nw, t+2nw may remain
        } else if (t + nw < S) {
            __builtin_amdgcn_s_wait_asynccnt(2);  // row t+nw may remain
        } else {
            __builtin_amdgcn_s_wait_asynccnt(0);
        }
        a = *(const v4f*)(&stage[wib][pb][lane * 8]);
        b = *(const v4f*)(&stage[wib][pb][lane * 8 + 4]);
        pb = (pb + 1) % NBUF;
#else
        const v4f* src = (const v4f*)(logits + (size_t)t * N_EXPERTS);
        a = __builtin_nontemporal_load(src + lane * 2);
        b = __builtin_nontemporal_load(src + lane * 2 + 1);
#endif
        float x[8] = {a.x, a.y, a.z, a.w, b.x, b.y, b.z, b.w};
        const int base = lane * 8;

        // lane-local argmax over its 8 elements (all in group lane/4);
        // strict '>' keeps lowest index on ties, matching jax.lax.top_k
        float mv = x[0];
        int   mi = base;
#pragma unroll
        for (int j = 1; j < 8; ++j)
            if (x[j] > mv) { mv = x[j]; mi = base + j; }

        // quad reduction -> per-group (32 experts) argmax, on all 4 quad lanes
#pragma unroll
        for (int d = 1; d <= 2; d <<= 1) {
            float ov = __shfl_xor(mv, d, 32);
            int   oi = __shfl_xor(mi, d, 32);
            if (ov > mv || (ov == mv && oi < mi)) { mv = ov; mi = oi; }
        }
        const float gval = mv;
        const int   gidx = mi;  // already a global expert id in [0,256)

        // row max = max of group maxes (cross-quad levels only)
        float rmax = gval;
#pragma unroll
        for (int d = 4; d <= 16; d <<= 1)
            rmax = fmaxf(rmax, __shfl_xor(rmax, d, 32));

        // exp and wave-wide sum for softmax
        float e[8];
        float ssum = 0.0f;
#pragma unroll
        for (int j = 0; j < 8; ++j) { e[j] = __expf(x[j] - rmax); ssum += e[j]; }
#pragma unroll
        for (int d = 1; d <= 16; d <<= 1)
            ssum += __shfl_xor(ssum, d, 32);
        const float inv = 1.0f / ssum;

        // streaming outputs: logits passthrough + routing weights (NT stores)
        const size_t rowoff = (size_t)t * N_EXPERTS;
        v4f r0 = {e[0] * inv, e[1] * inv, e[2] * inv, e[3] * inv};
        v4f r1 = {e[4] * inv, e[5] * inv, e[6] * inv, e[7] * inv};
        __builtin_nontemporal_store(a,  (v4f*)(out_logits + rowoff) + lane * 2);
        __builtin_nontemporal_store(b,  (v4f*)(out_logits + rowoff) + lane * 2 + 1);
        __builtin_nontemporal_store(r0, (v4f*)(out_rw + rowoff) + lane * 2);
        __builtin_nontemporal_store(r1, (v4f*)(out_rw + rowoff) + lane * 2 + 1);

        // top-1 per group: every lane in a quad already holds its group's gexp,
        // so summing across quads (xor 4,8,16) yields sum over the 8 winners.
        const float gexp = __expf(gval - rmax);
        float gsum = gexp;
#pragma unroll
        for (int d = 4; d <= 16; d <<= 1)
            gsum += __shfl_xor(gsum, d, 32);
        const float tw = (gexp * inv) / (gsum * inv + 1e-20f);

        if ((lane & 3) == 0) {
            const int g = lane >> 2;  // group id 0..7, reference order
            out_tw [(size_t)t * TOP_K + g] = tw;
            out_tid[(size_t)t * TOP_K + g] = (float)gidx;
            atomicAdd(&hist[gidx], 1u);
        }
    }

    __syncthreads();
    // drain block-local histogram: exact + order-independent (counts < 2^24)
    for (int i = threadIdx.x; i < N_EXPERTS; i += THREADS) {
        unsigned int c = hist[i];
        if (c) atomicAdd(&out_hist[i], (float)c);
    }
}

extern "C" void kernel_launch(void* const* d_in, const int* in_sizes, int n_in,
                              void* d_out, int out_size, void* d_ws, size_t ws_size,
                              hipStream_t stream) {
    const float* logits = (const float*)d_in[0];
    const int S = in_sizes[0] / N_EXPERTS;

    float* out        = (float*)d_out;
    float* out_logits = out;
    float* out_rw     = out_logits + (size_t)S * N_EXPERTS;
    float* out_tw     = out_rw     + (size_t)S * N_EXPERTS;
    float* out_tid    = out_tw     + (size_t)S * TOP_K;
    float* out_hist   = out_tid    + (size_t)S * TOP_K;

    zero_hist_kernel<<<1, N_EXPERTS, 0, stream>>>(out_hist);
    router_kernel<<<NBLOCKS, THREADS, 0, stream>>>(
        logits, S, out_logits, out_rw, out_tw, out_tid, out_hist);
}